// Measurements_80470507257900
// MI455X (gfx1250) — compile-verified
//
#include <hip/hip_runtime.h>
#include <stdint.h>

// ---------------------------------------------------------------------------
// Problem constants (from the reference):
//   in:  (8,4,16,256,256) f32, weight: (16,8,8) f32
//   out: (8, 4096, 256) f32 ; out[b, s*1024+i*32+j, q*16+p] = red[b,s,8i+p,8j+q]
//   red[b,s,h,w] = sum_t in[b,s,t,h,w] * weight[t, h%8, w%8]
// Memory-bound: 128MB read + 32MB write ~= 6.9us @ 23.3 TB/s. FLOPs ~67M.
// ---------------------------------------------------------------------------

typedef __attribute__((ext_vector_type(2))) float        v2f;
typedef __attribute__((ext_vector_type(4))) float        f4v;
typedef __attribute__((ext_vector_type(8))) float        v8f;
typedef __attribute__((ext_vector_type(4))) unsigned int v4u;
typedef __attribute__((ext_vector_type(4))) int          v4i;
typedef __attribute__((ext_vector_type(8))) int          v8i;

#define T_DIM   16
#define H_DIM   256
#define W_DIM   256
#define TILE_H  16
#define TILE_W  128   // must be >= 128: WMMA pixel group spans c + 8*15 <= 127

// ---------------------------------------------------------------------------
// Fused kernel: one workgroup = (bs, hblock, whalf), owns h in [h0,h0+16),
// w in [w0,w0+128). Stages 16t x 16h x 128w f32 (128KB) into LDS via TDM,
// then 8 waves (wave id = c = w%8 residue) each run 16 pixel-groups:
//   group (lh): 16 pixels w = w0 + c + 8n, n=0..15, fixed h = h0+lh.
//   D = A(16x4 weights, rows identical) x B(4x16 pixels), chained over 4
//   K-chunks -> sum over t=0..15. acc[0] in lanes 0..15 = pixel results.
// Scatter each pixel to its <=4 unique output patch slots.
// ---------------------------------------------------------------------------
__global__ __launch_bounds__(256) void fused_wmma_kernel(
    const float* __restrict__ in, const float* __restrict__ wt,
    float* __restrict__ out) {
  __shared__ __align__(16) float lds_in[T_DIM * TILE_H * TILE_W]; // 128 KB
  __shared__ __align__(16) float lds_w[T_DIM * 64];               // 4 KB

  const int tid = threadIdx.x;
  const int wg  = blockIdx.x;
  const int wh  = wg & 1;          // which 128-wide half of the row
  const int hb  = (wg >> 1) & 15;  // h block
  const int bs  = wg >> 5;         // fused (b*4+s), 0..31
  const int h0  = hb * TILE_H;
  const int w0  = wh * TILE_W;

  // Stage weights (1024 floats) into LDS: 256 threads x float4, coalesced.
  ((f4v*)lds_w)[tid] = ((const f4v*)wt)[tid];

#if __has_builtin(__builtin_amdgcn_tensor_load_to_lds)
  // --- Tensor Data Mover: one descriptor DMAs the whole 128KB tile ---------
  {
    const int wv = __builtin_amdgcn_readfirstlane(tid >> 5);
    if (wv == 0) {  // scalar branch: only wave 0 issues the TDM op
      const unsigned long long ga =
          (unsigned long long)(uintptr_t)in +
          4ull * ((unsigned long long)bs * T_DIM * H_DIM * W_DIM +
                  (unsigned long long)h0 * W_DIM + (unsigned long long)w0);
      const unsigned int la = (unsigned int)(uintptr_t)(&lds_in[0]);

      v4u g0;
      g0[0] = 1u;                                   // count=1, gather off
      g0[1] = la;                                   // lds_addr (bytes)
      g0[2] = (unsigned int)ga;                     // global_addr[31:0]
      g0[3] = (unsigned int)((ga >> 32) & 0x1FFFFFFu) | (2u << 30); // type=2

      v8i g1;
      g1[0] = (int)(2u << 16);                      // data_size = 4B
      g1[1] = (int)((unsigned)TILE_W << 16);        // tensor_dim0 lo16 (=tile)
      g1[2] = (int)((unsigned)TILE_H << 16);        // tensor_dim1 lo16
      g1[3] = (int)((unsigned)TILE_W << 16);        // tile_dim0 = 128 (w)
      g1[4] = (int)(TILE_H | (T_DIM << 16));        // tile_dim1=16(h), tile_dim2=16(t)
      g1[5] = W_DIM;                                // tensor_dim0_stride = 256
      g1[6] = 0;                                    // stride0 hi | stride1 lo16 (65536&0xFFFF=0)
      g1[7] = (int)((unsigned)(H_DIM * W_DIM) >> 16); // stride1[47:16] = 1

      v4i g2; g2[0] = T_DIM; g2[1] = 0; g2[2] = 0; g2[3] = 0; // tensor_dim2=16
      v4i g3; g3[0] = 0; g3[1] = 0; g3[2] = 0; g3[3] = 0;

#if defined(__clang_major__) && (__clang_major__ >= 23)
      v8i g4 = {0, 0, 0, 0, 0, 0, 0, 0};
      __builtin_amdgcn_tensor_load_to_lds(g0, g1, g2, g3, g4, 0);
#else
      __builtin_amdgcn_tensor_load_to_lds(g0, g1, g2, g3, 0);
#endif
      __builtin_amdgcn_s_wait_tensorcnt(0);
    }
  }
#else
  // --- Fallback: coalesced global float4 loads -> LDS ----------------------
  for (int k = 0; k < 32; ++k) {
    const int f  = tid + k * 256;      // float4 index in [0, 8192)
    const int wq = f & 31;             // 32 float4 per row
    const int lh = (f >> 5) & 15;
    const int t  = f >> 9;
    f4v v = *(const f4v*)(in + ((size_t)(bs * T_DIM + t) * H_DIM + (h0 + lh)) * W_DIM
                             + w0 + wq * 4);
    *(f4v*)(lds_in + (t * TILE_H + lh) * TILE_W + wq * 4) = v;
  }
#endif
  __syncthreads();

  const int lane = tid & 31;
  const int lo   = lane & 15;   // pixel index within group (B column N)
  const int hi   = lane >> 4;   // lane half selects K pair
  const int c    = tid >> 5;    // wave id == w%8 residue class, 0..7
  const int wl   = c + 8 * lo;  // local w in [0,128)

  for (int lh = 0; lh < 16; ++lh) {
    const int r = lh & 7;       // h%8 (h0 is a multiple of 16)
    v8f acc = {0.f, 0.f, 0.f, 0.f, 0.f, 0.f, 0.f, 0.f};
#pragma unroll
    for (int kc = 0; kc < 4; ++kc) {
      // Symmetric A/B per-lane K mapping (ISA 7.12.2, 32-bit 16x4 A table):
      // lanes 0-15: VGPR0=K0,VGPR1=K1 ; lanes 16-31: VGPR0=K2,VGPR1=K3.
      const int tt = 4 * kc + 2 * hi;
      v2f a, b;
      a.x = lds_w[tt * 64 + r * 8 + c];             // weight[tt,   r, c]
      a.y = lds_w[(tt + 1) * 64 + r * 8 + c];       // weight[tt+1, r, c]
      b.x = lds_in[(tt * TILE_H + lh) * TILE_W + wl];        // in[tt,   h, w]
      b.y = lds_in[((tt + 1) * TILE_H + lh) * TILE_W + wl];  // in[tt+1, h, w]
      acc = __builtin_amdgcn_wmma_f32_16x16x4_f32(
          false, a, false, b, (short)0, acc, false, false);
    }
    const float res = acc[0];   // lane(hi=0,lo=n): D[M=0][N=n], rows identical
    if (hi == 0) {
      const int h  = h0 + lh;
      const int w  = w0 + wl;
      const int i0 = h >> 3, p0 = h & 7;
      const int j0 = w >> 3, q0 = w & 7;
      const size_t base = (size_t)bs * (1024 * 256);
      // out flat = (bs*1024 + i*32 + j)*256 + q*16 + p ; each slot unique.
      out[base + (size_t)((i0 * 32 + j0) * 256) + q0 * 16 + p0] = res;
      if (j0 > 0)
        out[base + (size_t)((i0 * 32 + j0 - 1) * 256) + (q0 + 8) * 16 + p0] = res;
      if (i0 > 0)
        out[base + (size_t)(((i0 - 1) * 32 + j0) * 256) + q0 * 16 + (p0 + 8)] = res;
      if (i0 > 0 && j0 > 0)
        out[base + (size_t)(((i0 - 1) * 32 + j0 - 1) * 256) + (q0 + 8) * 16 + (p0 + 8)] = res;
    }
  }
}

// ---------------------------------------------------------------------------
// Zero the pad-border output elements (h=8i+p>=256 or w=8j+q>=256), i.e.
// (i==31 && p>=8) or (j==31 && q>=8). 8192 elems per (b,s), 262144 total.
// Overlap region is double-written with 0 (harmless).
// ---------------------------------------------------------------------------
__global__ __launch_bounds__(256) void border_zero_kernel(float* __restrict__ out) {
  const int z    = blockIdx.x * 256 + threadIdx.x;  // [0, 262144)
  const int bs   = z >> 13;                         // 32 (b,s) pairs
  const int r    = z & 8191;
  const int half = r >> 12;
  const int rr   = r & 4095;
  int i, j, p, q;
  if (half == 0) { i = 31; j = rr >> 7; q = (rr >> 3) & 15; p = 8 + (rr & 7); }
  else           { j = 31; i = rr >> 7; p = (rr >> 3) & 15; q = 8 + (rr & 7); }
  out[(size_t)bs * (1024 * 256) + (size_t)((i * 32 + j) * 256) + q * 16 + p] = 0.0f;
}

extern "C" void kernel_launch(void* const* d_in, const int* in_sizes, int n_in,
                              void* d_out, int out_size, void* d_ws, size_t ws_size,
                              hipStream_t stream) {
  (void)in_sizes; (void)n_in; (void)out_size; (void)d_ws; (void)ws_size;
  const float* in = (const float*)d_in[0];
  const float* wt = (const float*)d_in[1];
  float* out = (float*)d_out;

  // Disjoint output regions -> order irrelevant; same stream serializes anyway.
  hipLaunchKernelGGL(border_zero_kernel, dim3(1024), dim3(256), 0, stream, out);
  // 32 (b,s) x 16 h-blocks x 2 w-halves = 1024 workgroups, 256 threads each.
  hipLaunchKernelGGL(fused_wmma_kernel, dim3(1024), dim3(256), 0, stream, in, wt, out);
}